// KMedoids_70600672412343
// MI455X (gfx1250) — compile-verified
//
#include <hip/hip_runtime.h>

// Problem constants (match reference)
#define BB 16
#define NN 2048
#define CC 256
#define KK 128
#define ITERS 5
#define LDS_STRIDE 260   // 16-row A tile, padded to spread rows across 64 LDS banks

typedef float v2f __attribute__((ext_vector_type(2)));
typedef float v8f __attribute__((ext_vector_type(8)));

// ---------------------------------------------------------------------------
// sq[b,i] = sum_c x[b,i,c]^2   (one wave32 per row)
// ---------------------------------------------------------------------------
__global__ void kmed_sq_kernel(const float* __restrict__ x, float* __restrict__ sq) {
    int row  = blockIdx.x * 8 + (threadIdx.x >> 5);   // [0, B*N)
    int lane = threadIdx.x & 31;
    const float* xr = x + (size_t)row * CC;
    float s = 0.f;
    #pragma unroll
    for (int c = lane; c < CC; c += 32) { float v = xr[c]; s += v * v; }
    #pragma unroll
    for (int off = 16; off >= 1; off >>= 1) s += __shfl_xor(s, off, 32);
    if (lane == 0) sq[row] = s;
}

// ---------------------------------------------------------------------------
// top-K init via rank counting: stable like jax.lax.top_k
// (value descending, index ascending on ties)
// ---------------------------------------------------------------------------
__global__ void kmed_topk_kernel(const float* __restrict__ tw, int* __restrict__ cidx) {
    int b = blockIdx.x;
    __shared__ unsigned int wbits[NN];
    const float* wb = tw + (size_t)b * NN;
    for (int i = threadIdx.x; i < NN; i += blockDim.x)
        wbits[i] = __float_as_uint(wb[i]);            // weights >= 0: bits order == value order
    __syncthreads();
    for (int i = threadIdx.x; i < NN; i += blockDim.x) {
        unsigned int vi = wbits[i];
        int rank = 0;
        for (int j = 0; j < NN; ++j) {
            unsigned int vj = wbits[j];
            rank += (vj > vi) || (vj == vi && j < i);
        }
        if (rank < KK) cidx[b * KK + rank] = i;
    }
}

// ---------------------------------------------------------------------------
// wrow[b,i] = tw[b,i] * sum_j dist(i,j).
// Block = 8 waves; block owns a 16-row I tile (staged in LDS); wave w covers
// J tiles {w, w+8, ...}. Gram via V_WMMA_F32_16X16X4_F32, K-loop over C=256.
// ---------------------------------------------------------------------------
__global__ void kmed_wrow_kernel(const float* __restrict__ x, const float* __restrict__ tw,
                                 const float* __restrict__ sq, float* __restrict__ wrow) {
    __shared__ float xa[16 * LDS_STRIDE];
    __shared__ float partial[8][16];
    int b  = blockIdx.y;
    int i0 = blockIdx.x * 16;
    int tid = threadIdx.x, wave = tid >> 5, lane = tid & 31;
    const float* xb = x + (size_t)b * NN * CC;

    // stage A tile: 16 rows x 256 floats
    for (int e = tid; e < 16 * CC; e += 256) {
        int r = e >> 8, c = e & 255;
        xa[r * LDS_STRIDE + c] = xb[(size_t)(i0 + r) * CC + c];
    }
    __syncthreads();

    int hrow  = lane & 15;
    int koff  = (lane >> 4) * 2;       // lanes 16-31 hold K=k0+2,k0+3
    int mbase = (lane >> 4) * 8;       // D-frag: vgpr r -> row r (lanes<16) / r+8 (lanes>=16)
    float sqi[8];
    #pragma unroll
    for (int r = 0; r < 8; ++r) sqi[r] = sq[b * NN + i0 + mbase + r];

    float acc_row[8];
    #pragma unroll
    for (int r = 0; r < 8; ++r) acc_row[r] = 0.f;

    const float* ap = xa + hrow * LDS_STRIDE + koff;
    for (int jt = wave; jt < NN / 16; jt += 8) {
        int j0 = jt * 16;
        float sqj = sq[b * NN + j0 + hrow];
        const float* bp = xb + (size_t)(j0 + hrow) * CC + koff;
        v8f acc = {};
        #pragma unroll 8
        for (int k0 = 0; k0 < CC; k0 += 4) {
            v2f a  = *(const v2f*)(ap + k0);
            v2f bv = *(const v2f*)(bp + k0);
            acc = __builtin_amdgcn_wmma_f32_16x16x4_f32(false, a, false, bv,
                                                        (short)0, acc, false, false);
        }
        #pragma unroll
        for (int r = 0; r < 8; ++r) {
            float d2 = sqi[r] + sqj - 2.f * acc[r];
            d2 = fmaxf(d2, 0.f);
            acc_row[r] += (d2 > 0.f) ? sqrtf(d2) : 0.f;
        }
    }
    // reduce the 16 columns (lanes) of each half-wave
    #pragma unroll
    for (int r = 0; r < 8; ++r) {
        float s = acc_row[r];
        s += __shfl_xor(s, 1, 32);
        s += __shfl_xor(s, 2, 32);
        s += __shfl_xor(s, 4, 32);
        s += __shfl_xor(s, 8, 32);
        if ((lane & 15) == 0) partial[wave][mbase + r] = s;
    }
    __syncthreads();
    if (tid < 16) {
        float tot = 0.f;
        #pragma unroll
        for (int wv = 0; wv < 8; ++wv) tot += partial[wv][tid];
        int gi = b * NN + i0 + tid;
        wrow[gi] = tw[gi] * tot;
    }
}

// ---------------------------------------------------------------------------
// assignment[b,i] = argmin_k dist(i, medoid_k), tie -> smallest k.
// Block = 8 waves; wave handles 16 rows x all 128 centers (8 WMMA acc tiles).
// ---------------------------------------------------------------------------
__global__ void kmed_assign_kernel(const float* __restrict__ x, const float* __restrict__ sq,
                                   const int* __restrict__ cidx, int* __restrict__ assign) {
    int b = blockIdx.y;
    int wave = threadIdx.x >> 5, lane = threadIdx.x & 31;
    int i0 = blockIdx.x * 128 + wave * 16;
    int hrow = lane & 15, koff = (lane >> 4) * 2, mbase = (lane >> 4) * 8;
    const float* xb = x + (size_t)b * NN * CC;
    const int*   cb = cidx + b * KK;

    int   cj[8];
    float sqc[8];
    #pragma unroll
    for (int t = 0; t < 8; ++t) {
        cj[t]  = cb[t * 16 + hrow];
        sqc[t] = sq[b * NN + cj[t]];
    }
    float sqi[8];
    #pragma unroll
    for (int r = 0; r < 8; ++r) sqi[r] = sq[b * NN + i0 + mbase + r];

    v8f zero = {};
    v8f acc[8];
    #pragma unroll
    for (int t = 0; t < 8; ++t) acc[t] = zero;

    const float* ap = xb + (size_t)(i0 + hrow) * CC + koff;
    for (int k0 = 0; k0 < CC; k0 += 4) {
        v2f a = *(const v2f*)(ap + k0);
        #pragma unroll
        for (int t = 0; t < 8; ++t) {
            v2f bv = *(const v2f*)(xb + (size_t)cj[t] * CC + koff + k0);
            acc[t] = __builtin_amdgcn_wmma_f32_16x16x4_f32(false, a, false, bv,
                                                           (short)0, acc[t], false, false);
        }
    }

    float minv[8]; int mini[8];
    #pragma unroll
    for (int r = 0; r < 8; ++r) { minv[r] = 3.4e38f; mini[r] = 0; }
    #pragma unroll
    for (int t = 0; t < 8; ++t) {
        int kidx = t * 16 + hrow;
        #pragma unroll
        for (int r = 0; r < 8; ++r) {
            float d2 = sqi[r] + sqc[t] - 2.f * acc[t][r];
            d2 = fmaxf(d2, 0.f);
            float d = (d2 > 0.f) ? sqrtf(d2) : 0.f;
            if (d < minv[r] || (d == minv[r] && kidx < mini[r])) { minv[r] = d; mini[r] = kidx; }
        }
    }
    #pragma unroll
    for (int r = 0; r < 8; ++r) {
        float v = minv[r]; int idx = mini[r];
        #pragma unroll
        for (int off = 1; off < 16; off <<= 1) {
            float ov = __shfl_xor(v, off, 32);
            int   oi = __shfl_xor(idx, off, 32);
            if (ov < v || (ov == v && oi < idx)) { v = ov; idx = oi; }
        }
        if ((lane & 15) == 0) assign[b * NN + i0 + mbase + r] = idx;
    }
}

// ---------------------------------------------------------------------------
// cluster_idx[b,k] = argmin_i (assignment[b,i]==k ? wrow[b,i] : big), tie -> smallest i,
// empty cluster -> 0 (matches argmin over all-`big` column). One wave per (b,k).
// ---------------------------------------------------------------------------
__global__ void kmed_update_kernel(const float* __restrict__ wrow, const int* __restrict__ assign,
                                   int* __restrict__ cidx) {
    int gw = blockIdx.x * 8 + (threadIdx.x >> 5);
    int b = gw >> 7, k = gw & 127;
    int lane = threadIdx.x & 31;
    const float* wb = wrow + (size_t)b * NN;
    const int*   ab = assign + (size_t)b * NN;
    float minv = 2.048e9f;   // N * 1e6, exactly representable
    int   mini = 0;
    for (int i = lane; i < NN; i += 32) {
        if (ab[i] == k) {
            float v = wb[i];
            if (v < minv || (v == minv && i < mini)) { minv = v; mini = i; }
        }
    }
    #pragma unroll
    for (int off = 1; off < 32; off <<= 1) {
        float ov = __shfl_xor(minv, off, 32);
        int   oi = __shfl_xor(mini, off, 32);
        if (ov < minv || (ov == minv && oi < mini)) { minv = ov; mini = oi; }
    }
    if (lane == 0) cidx[b * KK + k] = mini;
}

// ---------------------------------------------------------------------------
// outputs: centers gather + cluster_idx/assignment as floats
// ---------------------------------------------------------------------------
__global__ void kmed_output_kernel(const float* __restrict__ x, const int* __restrict__ cidx,
                                   float* __restrict__ out_centers, float* __restrict__ out_cidx) {
    int b = blockIdx.y, k = blockIdx.x;
    int ci = cidx[b * KK + k];
    const float* src = x + ((size_t)b * NN + ci) * CC;
    float* dst = out_centers + ((size_t)b * KK + k) * CC;
    dst[threadIdx.x] = src[threadIdx.x];
    if (threadIdx.x == 0) out_cidx[b * KK + k] = (float)ci;
}

__global__ void kmed_assign_out_kernel(const int* __restrict__ assign, float* __restrict__ out) {
    int i = blockIdx.x * 256 + threadIdx.x;
    out[i] = (float)assign[i];
}

// ---------------------------------------------------------------------------
extern "C" void kernel_launch(void* const* d_in, const int* in_sizes, int n_in,
                              void* d_out, int out_size, void* d_ws, size_t ws_size,
                              hipStream_t stream) {
    (void)in_sizes; (void)n_in; (void)out_size; (void)ws_size;
    const float* x  = (const float*)d_in[0];   // (B, N, C)
    const float* tw = (const float*)d_in[1];   // (B, N, 1)

    float* ws     = (float*)d_ws;
    float* sq     = ws;                         // B*N
    float* wrow   = ws + BB * NN;               // B*N
    int*   cidx   = (int*)(ws + 2 * BB * NN);   // B*K
    int*   assign = cidx + BB * KK;             // B*N

    float* out_centers = (float*)d_out;                 // B*K*C
    float* out_cidx    = out_centers + BB * KK * CC;    // B*K
    float* out_assign  = out_cidx + BB * KK;            // B*N

    kmed_sq_kernel   <<<BB * NN / 8, 256, 0, stream>>>(x, sq);
    kmed_topk_kernel <<<BB,          256, 0, stream>>>(tw, cidx);
    kmed_wrow_kernel <<<dim3(NN / 16, BB), 256, 0, stream>>>(x, tw, sq, wrow);

    for (int it = 0; it < ITERS; ++it) {
        kmed_assign_kernel<<<dim3(NN / 128, BB), 256, 0, stream>>>(x, sq, cidx, assign);
        kmed_update_kernel<<<BB * KK / 8, 256, 0, stream>>>(wrow, assign, cidx);
    }
    kmed_assign_kernel<<<dim3(NN / 128, BB), 256, 0, stream>>>(x, sq, cidx, assign);

    kmed_output_kernel    <<<dim3(KK, BB), CC, 0, stream>>>(x, cidx, out_centers, out_cidx);
    kmed_assign_out_kernel<<<BB * NN / 256, 256, 0, stream>>>(assign, out_assign);
}